// PREF_18021682774691
// MI455X (gfx1250) — compile-verified
//
#include <hip/hip_runtime.h>
#include <stdint.h>

#define NCHN 16          // CH
#define NR   7           // R
#define RES  128
#define NCH  (2*NCHN*NR) // 224 channels per plane
#define NPIX (RES*RES)   // 16384 pixels

// ---------------------------------------------------------------------------
// Pass 1: relayout each plane from (C, H, W) to (H, W, C) so a bilinear tap
// reads one contiguous 896B vector (global_load_b128 stream, L2-friendly).
// ---------------------------------------------------------------------------
__global__ __launch_bounds__(256) void relayout_kernel(
    const float* __restrict__ Pu, const float* __restrict__ Pv,
    const float* __restrict__ Pw, float* __restrict__ out) {
  __shared__ float tile[32][33];
  const float* src = (blockIdx.z == 0) ? Pu : ((blockIdx.z == 1) ? Pv : Pw);
  float* dst = out + (size_t)blockIdx.z * (size_t)NCH * NPIX;
  const int p0 = blockIdx.x * 32;   // pixel tile
  const int c0 = blockIdx.y * 32;   // channel tile
  const int tx = threadIdx.x;       // 0..31
  const int ty = threadIdx.y;       // 0..7
#pragma unroll
  for (int i = ty; i < 32; i += 8)
    tile[i][tx] = src[(size_t)(c0 + i) * NPIX + (p0 + tx)];  // coalesced over pixels
  __syncthreads();
#pragma unroll
  for (int i = ty; i < 32; i += 8)
    dst[(size_t)(p0 + i) * NCH + (c0 + tx)] = tile[tx][i];   // coalesced over channels
}

// ---------------------------------------------------------------------------
// Pass 2: one thread per sample. Stage the block's coordinate slice into LDS
// with CDNA5 async global->LDS copies, then gather 4 taps per axis and fold
// them against the per-sample Fourier basis.
// ---------------------------------------------------------------------------
template <bool TRANSPOSED>
__global__ __launch_bounds__(256) void fourier_kernel(
    const float* __restrict__ inputs,
    const float* __restrict__ Pu, const float* __restrict__ Pv,
    const float* __restrict__ Pw, float* __restrict__ out, int Mtot) {
  __shared__ float s_in[768];  // 256 samples * 3 coords
  const int t = threadIdx.x;
  const size_t base = (size_t)blockIdx.x * 768;

  // Async stage: 3072 bytes, 192 lanes x b128 (tracked with ASYNCcnt).
  if (t < 192 && (base + (size_t)(t + 1) * 4) <= (size_t)Mtot * 3) {
    const float* g = inputs + base + (size_t)t * 4;
    uint32_t lds_off = (uint32_t)(size_t)(&s_in[t * 4]);
    asm volatile("global_load_async_to_lds_b128 %0, %1, off"
                 :: "v"(lds_off), "v"(g) : "memory");
  }
  asm volatile("s_wait_asynccnt 0x0" ::: "memory");
  __syncthreads();

  const size_t m = (size_t)blockIdx.x * 256 + t;
  if (m >= (size_t)Mtot) return;

  const float in0 = s_in[t * 3 + 0];
  const float in1 = s_in[t * 3 + 1];
  const float in2 = s_in[t * 3 + 2];

  float acc[NCHN];
#pragma unroll
  for (int s = 0; s < NCHN; ++s) acc[s] = 0.0f;

#pragma unroll 1
  for (int axis = 0; axis < 3; ++axis) {
    // axis 0 (u): gx=in1 gy=in2 coord=in0 ; axis 1 (v): gx=in0 gy=in2 coord=in1
    // axis 2 (w): gx=in0 gy=in1 coord=in2
    const float gx = (axis == 0) ? in1 : in0;
    const float gy = (axis == 2) ? in1 : in2;
    const float co = (axis == 0) ? in0 : ((axis == 1) ? in1 : in2);
    const float* plane = (axis == 0) ? Pu : ((axis == 1) ? Pv : Pw);

    // grid_sample, align_corners=True
    const float xf = (gx + 1.0f) * 0.5f * (float)(RES - 1);
    const float yf = (gy + 1.0f) * 0.5f * (float)(RES - 1);
    const float x0f = floorf(xf), y0f = floorf(yf);
    const float wx = xf - x0f, wy = yf - y0f;
    int x0 = min(max((int)x0f, 0), RES - 1);
    int y0 = min(max((int)y0f, 0), RES - 1);
    const int x1 = min(x0 + 1, RES - 1);
    const int y1 = min(y0 + 1, RES - 1);
    const float w00 = (1.0f - wx) * (1.0f - wy);
    const float w01 = wx * (1.0f - wy);
    const float w10 = (1.0f - wx) * wy;
    const float w11 = wx * wy;

    const float* p00;
    const float* p01;
    const float* p10;
    const float* p11;
    if constexpr (TRANSPOSED) {
      p00 = plane + (size_t)(y0 * RES + x0) * NCH;
      p01 = plane + (size_t)(y0 * RES + x1) * NCH;
      p10 = plane + (size_t)(y1 * RES + x0) * NCH;
      p11 = plane + (size_t)(y1 * RES + x1) * NCH;
      // kick the tap lines toward the WGP while we do the sincos chain
      __builtin_prefetch(p00, 0, 0); __builtin_prefetch(p00 + 128, 0, 0);
      __builtin_prefetch(p01, 0, 0); __builtin_prefetch(p01 + 128, 0, 0);
      __builtin_prefetch(p10, 0, 0); __builtin_prefetch(p10 + 128, 0, 0);
      __builtin_prefetch(p11, 0, 0); __builtin_prefetch(p11 + 128, 0, 0);
    } else {
      p00 = plane + (size_t)(y0 * RES + x0);
      p01 = plane + (size_t)(y0 * RES + x1);
      p10 = plane + (size_t)(y1 * RES + x0);
      p11 = plane + (size_t)(y1 * RES + x1);
    }

    // Fourier basis: phase_r = (2*pi/128) * scaled * (2^r - 1)
    const float scaled = (co + 1.0f) * 0.5f * 127.0f;
    const float pbase = (6.283185307179586f / 128.0f) * scaled;
    const float coef[NR] = {0.f, 1.f, 3.f, 7.f, 15.f, 31.f, 63.f};
    float cs[NR], sn[NR];
#pragma unroll
    for (int r = 0; r < NR; ++r) __sincosf(pbase * coef[r], &sn[r], &cs[r]);

    if constexpr (TRANSPOSED) {
      const float4* q00 = reinterpret_cast<const float4*>(p00);
      const float4* q01 = reinterpret_cast<const float4*>(p01);
      const float4* q10 = reinterpret_cast<const float4*>(p10);
      const float4* q11 = reinterpret_cast<const float4*>(p11);
#pragma unroll
      for (int g = 0; g < NCH / 4; ++g) {
        const float4 a = q00[g];
        const float4 b = q01[g];
        const float4 c = q10[g];
        const float4 d = q11[g];
        const float fv[4] = {
            w00 * a.x + w01 * b.x + w10 * c.x + w11 * d.x,
            w00 * a.y + w01 * b.y + w10 * c.y + w11 * d.y,
            w00 * a.z + w01 * b.z + w10 * c.z + w11 * d.z,
            w00 * a.w + w01 * b.w + w10 * c.w + w11 * d.w};
#pragma unroll
        for (int j = 0; j < 4; ++j) {
          const int ch = g * 4 + j;      // c = s*7 + r (re), 112 + s*7 + r (im)
          const int r = ch % NR;
          if (ch < NCHN * NR)
            acc[ch / NR] = fmaf(cs[r], fv[j], acc[ch / NR]);
          else
            acc[(ch - NCHN * NR) / NR] =
                fmaf(-sn[r], fv[j], acc[(ch - NCHN * NR) / NR]);
        }
      }
    } else {
      // fallback: original (C,H,W) layout, channel stride NPIX
#pragma unroll
      for (int s = 0; s < NCHN; ++s) {
#pragma unroll
        for (int r = 0; r < NR; ++r) {
          const size_t ore = (size_t)(s * NR + r) * NPIX;
          const size_t oim = ore + (size_t)(NCHN * NR) * NPIX;
          const float fre =
              w00 * p00[ore] + w01 * p01[ore] + w10 * p10[ore] + w11 * p11[ore];
          const float fim =
              w00 * p00[oim] + w01 * p01[oim] + w10 * p10[oim] + w11 * p11[oim];
          acc[s] += cs[r] * fre - sn[r] * fim;
        }
      }
    }
  }

  float4* o = reinterpret_cast<float4*>(out + m * NCHN);  // 64B per sample
  o[0] = make_float4(acc[0], acc[1], acc[2], acc[3]);
  o[1] = make_float4(acc[4], acc[5], acc[6], acc[7]);
  o[2] = make_float4(acc[8], acc[9], acc[10], acc[11]);
  o[3] = make_float4(acc[12], acc[13], acc[14], acc[15]);
}

// ---------------------------------------------------------------------------
extern "C" void kernel_launch(void* const* d_in, const int* in_sizes, int n_in,
                              void* d_out, int out_size, void* d_ws,
                              size_t ws_size, hipStream_t stream) {
  const float* inputs = (const float*)d_in[0];
  const float* Pu = (const float*)d_in[1];
  const float* Pv = (const float*)d_in[2];
  const float* Pw = (const float*)d_in[3];
  float* out = (float*)d_out;

  const int Mtot = in_sizes[0] / 3;
  const int nblocks = (Mtot + 255) / 256;
  const size_t plane = (size_t)NCH * NPIX;
  const size_t need = 3 * plane * sizeof(float);

  if (ws_size >= need) {
    float* T = (float*)d_ws;
    relayout_kernel<<<dim3(NPIX / 32, NCH / 32, 3), dim3(32, 8), 0, stream>>>(
        Pu, Pv, Pw, T);
    fourier_kernel<true><<<nblocks, 256, 0, stream>>>(
        inputs, T, T + plane, T + 2 * plane, out, Mtot);
  } else {
    fourier_kernel<false><<<nblocks, 256, 0, stream>>>(
        inputs, Pu, Pv, Pw, out, Mtot);
  }
}